// WarpDTI_89197880803391
// MI455X (gfx1250) — compile-verified
//
#include <hip/hip_runtime.h>
#include <math.h>

// Problem geometry (fixed by the reference: B=1, C=6/3, 128^3).
constexpr int NX = 128, NY = 128, NZ = 128;
constexpr int NVOX = NX * NY * NZ;

// Block tile: 2 x 4 x 32 = 256 threads; wave = 32 consecutive z.
constexpr int TX = 2, TY = 4, TZ = 32;
constexpr int HTX = TX + 2, HTY = TY + 2, HTZ = TZ + 2;      // halo tile 4x6x34
constexpr int HALO_ROWS = 3 * HTX * HTY;                      // 72 z-rows
constexpr int SMEM_FLOATS = HALO_ROWS * HTZ;                  // 2448 (~9.6 KB)

#define SIDX(c, x, y, z) ((((c) * HTX + (x)) * HTY + (y)) * HTZ + (z))

#if __has_builtin(__builtin_amdgcn_global_load_async_to_lds_b32)
#define HAVE_ASYNC_LDS 1
#endif

typedef __attribute__((address_space(1))) int global_i32;
typedef __attribute__((address_space(3))) int lds_i32;

__device__ __forceinline__ void copy_g2lds(const float* g, float* l) {
#ifdef HAVE_ASYNC_LDS
  // gfx1250 async DMA: global -> LDS, tracked by ASYNCcnt.
  __builtin_amdgcn_global_load_async_to_lds_b32(
      (global_i32*)g, (lds_i32*)l, /*offset=*/0, /*cpol=*/0);
#else
  *l = *g;
#endif
}

__device__ __forceinline__ void wait_async_lds() {
#ifdef HAVE_ASYNC_LDS
#if __has_builtin(__builtin_amdgcn_s_wait_asynccnt)
  __builtin_amdgcn_s_wait_asynccnt(0);
#else
  asm volatile("s_wait_asynccnt 0" ::: "memory");
#endif
#endif
}

__global__ __launch_bounds__(256) void WarpDTI_fused_kernel(
    const float* __restrict__ dti,   // (6, 128,128,128)
    const float* __restrict__ ddf,   // (3, 128,128,128)
    float* __restrict__ out)         // (6, 128,128,128)
{
  __shared__ float s[SMEM_FLOATS];

  const int tid = threadIdx.x;
  const int lane = tid & 31;
  const int wv = tid >> 5;           // wave id 0..7
  const int z0 = blockIdx.x * TZ;
  const int y0 = blockIdx.y * TY;
  const int x0 = blockIdx.z * TX;

  // ---- Stage ddf halo tile (border-clamped) into LDS via async DMA.
  // 72 contiguous z-rows of 34 floats; wave w copies rows w, w+8, ...
  // lanes 0..31 cover z 0..31; lanes 0..1 cover the 2-element tail.
  {
    const int gz0 = min(max(z0 - 1 + lane, 0), NZ - 1);
    const int gz1 = min(z0 + 31 + lane, NZ - 1);     // z0-1+32+lane, >= 0 always
    for (int r = wv; r < HALO_ROWS; r += 8) {
      const int c  = r / 24;               // channel
      const int rr = r - c * 24;
      const int hx = rr / 6;
      const int hy = rr - hx * 6;
      const int gx = min(max(x0 - 1 + hx, 0), NX - 1);
      const int gy = min(max(y0 - 1 + hy, 0), NY - 1);
      const float* grow = ddf + (size_t)c * NVOX + (size_t)((gx * NY + gy) * NZ);
      float* lrow = &s[r * HTZ];
      copy_g2lds(grow + gz0, lrow + lane);
      if (lane < 2) copy_g2lds(grow + gz1, lrow + 32 + lane);
    }
  }
  wait_async_lds();
  __syncthreads();

  // ---- Per-voxel coordinates ----
  const int lz = lane;
  const int ly = wv & (TY - 1);
  const int lx = wv >> 2;
  const int x = x0 + lx, y = y0 + ly, z = z0 + lz;
  const int sx = lx + 1, sy = ly + 1, sz = lz + 1;

  const float d0 = s[SIDX(0, sx, sy, sz)];
  const float d1 = s[SIDX(1, sx, sy, sz)];
  const float d2 = s[SIDX(2, sx, sy, sz)];

  // ---- Warp coords, gather 8 corners x 6 channels, combine immediately
  // (short live ranges -> low VGPR pressure -> more waves to hide latency).
  const float cxf = fminf(fmaxf((float)x + d0, 0.0f), (float)(NX - 1));
  const float cyf = fminf(fmaxf((float)y + d1, 0.0f), (float)(NY - 1));
  const float czf = fminf(fmaxf((float)z + d2, 0.0f), (float)(NZ - 1));
  const float fxf = floorf(cxf), fyf = floorf(cyf), fzf = floorf(czf);
  const float fx = cxf - fxf, fy = cyf - fyf, fz = czf - fzf;
  const int ix0 = (int)fxf, iy0 = (int)fyf, iz0 = (int)fzf;
  const int ix1 = min(ix0 + 1, NX - 1);
  const int iy1 = min(iy0 + 1, NY - 1);
  const int iz1 = min(iz0 + 1, NZ - 1);

  const int b00 = (ix0 * NY + iy0) * NZ;
  const int b01 = (ix0 * NY + iy1) * NZ;
  const int b10 = (ix1 * NY + iy0) * NZ;
  const int b11 = (ix1 * NY + iy1) * NZ;

  const float ax0 = 1.0f - fx, ay0 = 1.0f - fy, az0 = 1.0f - fz;
  const float w000 = ax0 * ay0 * az0, w001 = ax0 * ay0 * fz;
  const float w010 = ax0 * fy * az0,  w011 = ax0 * fy * fz;
  const float w100 = fx * ay0 * az0,  w101 = fx * ay0 * fz;
  const float w110 = fx * fy * az0,   w111 = fx * fy * fz;

  float m[6];
#pragma unroll
  for (int c = 0; c < 6; ++c) {
    const float* p = dti + (size_t)c * NVOX;
    m[c] = w000 * p[b00 + iz0] + w001 * p[b00 + iz1]
         + w010 * p[b01 + iz0] + w011 * p[b01 + iz1]
         + w100 * p[b10 + iz0] + w101 * p[b10 + iz1]
         + w110 * p[b11 + iz0] + w111 * p[b11 + iz1];
  }

  // ---- Jacobian J = I + grad(ddf); clamped halo + edge scale 1.0
  // reproduces jnp.gradient one-sided differences at the borders.
  const float wxs = (x == 0 || x == NX - 1) ? 1.0f : 0.5f;
  const float wys = (y == 0 || y == NY - 1) ? 1.0f : 0.5f;
  const float wzs = (z == 0 || z == NZ - 1) ? 1.0f : 0.5f;

  float R[3][3];
#pragma unroll
  for (int c = 0; c < 3; ++c) {
    R[c][0] = (s[SIDX(c, sx + 1, sy, sz)] - s[SIDX(c, sx - 1, sy, sz)]) * wxs;
    R[c][1] = (s[SIDX(c, sx, sy + 1, sz)] - s[SIDX(c, sx, sy - 1, sz)]) * wys;
    R[c][2] = (s[SIDX(c, sx, sy, sz + 1)] - s[SIDX(c, sx, sy, sz - 1)]) * wzs;
  }
  R[0][0] += 1.0f; R[1][1] += 1.0f; R[2][2] += 1.0f;

  // ---- Polar orthogonal factor of J via det-scaled Newton:
  //   X <- 0.5*(zeta*X + (1/zeta)*X^{-T}),  zeta = |det X|^{-1/3}
  // Converges to U*Vh^T of the SVD (also for det<0) in <= 6 scaled steps.
#pragma unroll
  for (int it = 0; it < 6; ++it) {
    const float C00 = R[1][1] * R[2][2] - R[1][2] * R[2][1];
    const float C01 = R[1][2] * R[2][0] - R[1][0] * R[2][2];
    const float C02 = R[1][0] * R[2][1] - R[1][1] * R[2][0];
    const float C10 = R[0][2] * R[2][1] - R[0][1] * R[2][2];
    const float C11 = R[0][0] * R[2][2] - R[0][2] * R[2][0];
    const float C12 = R[0][1] * R[2][0] - R[0][0] * R[2][1];
    const float C20 = R[0][1] * R[1][2] - R[0][2] * R[1][1];
    const float C21 = R[0][2] * R[1][0] - R[0][0] * R[1][2];
    const float C22 = R[0][0] * R[1][1] - R[0][1] * R[1][0];
    const float det = R[0][0] * C00 + R[0][1] * C01 + R[0][2] * C02;
    const float ad  = fmaxf(fabsf(det), 1e-30f);
    const float zeta = exp2f(-(1.0f / 3.0f) * log2f(ad));
    const float dg = (fabsf(det) < 1e-30f) ? 1e-30f : det;  // sign-preserving guard
    const float a = 0.5f * zeta;
    const float b = 0.5f / (zeta * dg);                     // X^{-T} = C/det
    R[0][0] = a * R[0][0] + b * C00;
    R[0][1] = a * R[0][1] + b * C01;
    R[0][2] = a * R[0][2] + b * C02;
    R[1][0] = a * R[1][0] + b * C10;
    R[1][1] = a * R[1][1] + b * C11;
    R[1][2] = a * R[1][2] + b * C12;
    R[2][0] = a * R[2][0] + b * C20;
    R[2][1] = a * R[2][1] + b * C21;
    R[2][2] = a * R[2][2] + b * C22;
  }

  // ---- D' = R^T M R, M = [[m0,m1,m3],[m1,m2,m4],[m3,m4,m5]] ----
  const float M0[3] = {m[0], m[1], m[3]};
  const float M1[3] = {m[1], m[2], m[4]};
  const float M2[3] = {m[3], m[4], m[5]};

  float A[3][3];  // A = R^T M : A[i][j] = sum_k R[k][i]*M[k][j]
#pragma unroll
  for (int i = 0; i < 3; ++i) {
#pragma unroll
    for (int j = 0; j < 3; ++j) {
      A[i][j] = R[0][i] * M0[j] + R[1][i] * M1[j] + R[2][i] * M2[j];
    }
  }
  // Dp[i][j] = sum_k A[i][k]*R[k][j]; lower triangle only.
  const float D00 = A[0][0] * R[0][0] + A[0][1] * R[1][0] + A[0][2] * R[2][0];
  const float D10 = A[1][0] * R[0][0] + A[1][1] * R[1][0] + A[1][2] * R[2][0];
  const float D11 = A[1][0] * R[0][1] + A[1][1] * R[1][1] + A[1][2] * R[2][1];
  const float D20 = A[2][0] * R[0][0] + A[2][1] * R[1][0] + A[2][2] * R[2][0];
  const float D21 = A[2][0] * R[0][1] + A[2][1] * R[1][1] + A[2][2] * R[2][1];
  const float D22 = A[2][0] * R[0][2] + A[2][1] * R[1][2] + A[2][2] * R[2][2];

  const size_t lin = (size_t)((x * NY + y) * NZ + z);
  out[0 * (size_t)NVOX + lin] = D00;
  out[1 * (size_t)NVOX + lin] = D10;
  out[2 * (size_t)NVOX + lin] = D11;
  out[3 * (size_t)NVOX + lin] = D20;
  out[4 * (size_t)NVOX + lin] = D21;
  out[5 * (size_t)NVOX + lin] = D22;
}

extern "C" void kernel_launch(void* const* d_in, const int* in_sizes, int n_in,
                              void* d_out, int out_size, void* d_ws, size_t ws_size,
                              hipStream_t stream) {
  (void)in_sizes; (void)n_in; (void)out_size; (void)d_ws; (void)ws_size;
  const float* dti = (const float*)d_in[0];  // 6 * 128^3 floats
  const float* ddf = (const float*)d_in[1];  // 3 * 128^3 floats
  float* out = (float*)d_out;                // 6 * 128^3 floats

  dim3 grid(NZ / TZ, NY / TY, NX / TX);      // (4, 32, 64)
  WarpDTI_fused_kernel<<<grid, 256, 0, stream>>>(dti, ddf, out);
}